// adaptor_27711128994353
// MI455X (gfx1250) — compile-verified
//
#include <hip/hip_runtime.h>
#include <math.h>

typedef __attribute__((ext_vector_type(2))) float v2f;
typedef __attribute__((ext_vector_type(8))) float v8f;

#define OUT_HW   64
#define NPIX     4096      // 64*64
#define KDIM     1792
#define NCENT    4096
#define NNEI     200
#define NB       4
#define PCHUNK   512       // dist rows materialized at a time
#define KC       8         // K-chunk staged in LDS per iteration

// ---------------- avg_pool2d(3, stride=1, pad=1), count_include_pad -> /9 ----
__global__ void pool3_kernel(const float* __restrict__ x, float* __restrict__ y,
                             int C, int S) {
    int idx = blockIdx.x * blockDim.x + threadIdx.x;
    int total = NB * C * S * S;
    if (idx >= total) return;
    int w = idx % S;
    int h = (idx / S) % S;
    int bc = idx / (S * S);
    const float* xp = x + (size_t)bc * S * S;
    float s = 0.f;
    #pragma unroll
    for (int dh = -1; dh <= 1; ++dh) {
        int hh = h + dh;
        if (hh < 0 || hh >= S) continue;
        #pragma unroll
        for (int dw = -1; dw <= 1; ++dw) {
            int ww = w + dw;
            if (ww < 0 || ww >= S) continue;
            s += xp[hh * S + ww];
        }
    }
    y[idx] = s * (1.f / 9.f);
}

// ---------------- CoordConv 1x1: out[o,p] = sum_c w[o,c]*x[c,p] + coord terms + b
__global__ void coordconv_kernel(const float* __restrict__ pooled,
                                 const float* __restrict__ w,
                                 const float* __restrict__ bias,
                                 float* __restrict__ out,
                                 int Cin, int S) {
    int SS = S * S;
    int p = blockIdx.x * blockDim.x + threadIdx.x;
    if (p >= SS) return;
    int o = blockIdx.y;
    int b = blockIdx.z;
    int Cout = gridDim.y;
    const float* wp = w + (size_t)o * (Cin + 2);
    const float* xp = pooled + (size_t)b * Cin * SS + p;
    float acc = bias[o];
    for (int c = 0; c < Cin; ++c)
        acc += wp[c] * xp[(size_t)c * SS];
    int ww = p % S, hh = p / S;
    float xg = -1.f + 2.f * (float)ww / (float)(S - 1);
    float yg = -1.f + 2.f * (float)hh / (float)(S - 1);
    acc += wp[Cin] * xg + wp[Cin + 1] * yg;
    out[((size_t)b * Cout + o) * SS + p] = acc;
}

// ---------------- bilinear resize S->64 (half-pixel centers, edge clamp) -----
// writes into phi [b][chanOff+o][p]; channels < 896 also go to output#2
__global__ void resize_kernel(const float* __restrict__ src,
                              float* __restrict__ phi,
                              float* __restrict__ out2,
                              int S, int chanOff) {
    int p = blockIdx.x * blockDim.x + threadIdx.x;   // 0..4095
    if (p >= NPIX) return;
    int o = blockIdx.y;
    int b = blockIdx.z;
    int Cout = gridDim.y;
    int dx = p & 63, dy = p >> 6;
    float scale = (float)S / 64.f;
    float sx = ((float)dx + 0.5f) * scale - 0.5f;
    float sy = ((float)dy + 0.5f) * scale - 0.5f;
    sx = fminf(fmaxf(sx, 0.f), (float)(S - 1));
    sy = fminf(fmaxf(sy, 0.f), (float)(S - 1));
    int x0 = (int)sx; int x1 = min(x0 + 1, S - 1); float fx = sx - (float)x0;
    int y0 = (int)sy; int y1 = min(y0 + 1, S - 1); float fy = sy - (float)y0;
    const float* sp = src + ((size_t)b * Cout + o) * S * S;
    float v = (1.f - fy) * ((1.f - fx) * sp[y0 * S + x0] + fx * sp[y0 * S + x1])
            +        fy  * ((1.f - fx) * sp[y1 * S + x0] + fx * sp[y1 * S + x1]);
    int cg = chanOff + o;
    phi[((size_t)b * KDIM + cg) * NPIX + p] = v;
    if (cg < 896)
        out2[((size_t)b * 896 + cg) * NPIX + p] = v;
}

// ---------------- ||C_q||^2 ---------------------------------------------------
__global__ void cents_kernel(const float* __restrict__ Cm, float* __restrict__ cents) {
    int q = blockIdx.x * blockDim.x + threadIdx.x;
    if (q >= NCENT) return;
    float s = 0.f;
    for (int c = 0; c < KDIM; ++c) {
        float v = Cm[(size_t)c * NCENT + q];
        s += v * v;
    }
    cents[q] = s;
}

// ---------------- ||phi_{b,p}||^2 --------------------------------------------
__global__ void feats_kernel(const float* __restrict__ phi, float* __restrict__ feats) {
    int p = blockIdx.x * blockDim.x + threadIdx.x;
    if (p >= NPIX) return;
    int b = blockIdx.y;
    const float* pp = phi + (size_t)b * KDIM * NPIX + p;
    float s = 0.f;
    for (int c = 0; c < KDIM; ++c) {
        float v = pp[(size_t)c * NPIX];
        s += v * v;
    }
    feats[b * NPIX + p] = s;
}

// ---------------- fp32 WMMA GEMM + distance epilogue (LDS-staged) ------------
// D[p,q] = sum_c phi[c,p] * C[c,q]; dist = sqrt(feats[p]+cents[q]-2D)
// 128 threads = 4 waves per block; block tile 64(p) x 128(q);
// each wave computes a 32x64 quadrant (8 v8f accumulators).
__global__ void __launch_bounds__(128)
gemm_dist_kernel(const float* __restrict__ phiB, const float* __restrict__ Cm,
                 const float* __restrict__ featsB, const float* __restrict__ cents,
                 float* __restrict__ dist, int pc) {
    __shared__ float As[KC][64];    // [k][m] panel of phi^T
    __shared__ float Bs[KC][128];   // [k][n] panel of C

    int tid  = threadIdx.x;
    int lane = tid & 31;             // full wave32, EXEC all ones for WMMA
    int wid  = tid >> 5;             // 0..3
    int wm   = wid & 1;              // M half of block tile
    int wn   = wid >> 1;             // N half of block tile
    int half = lane >> 4;            // A/B fragment K-pair select
    int l16  = lane & 15;

    int p0 = pc + blockIdx.y * 64;   // block M base
    int q0 = blockIdx.x * 128;       // block N base

    // staging indices
    int akr = tid >> 4;              // 0..7   (A: one float4 per thread)
    int acm = (tid & 15) << 2;       // 0..60
    int bkr = tid >> 5;              // 0..3   (B: two float4 per thread)
    int bcn = (tid & 31) << 2;       // 0..124

    v8f zero = {};
    v8f acc[2][4];
    #pragma unroll
    for (int t = 0; t < 2; ++t)
        #pragma unroll
        for (int u = 0; u < 4; ++u) acc[t][u] = zero;

    int am0 = 32 * wm + l16;         // A fragment column (local m)
    int bn0 = 64 * wn + l16;         // B fragment column (local n)

    for (int kk = 0; kk < KDIM; kk += KC) {
        // ---- stage panels: global_load_b128 -> ds_store_b128
        *(float4*)&As[akr][acm] =
            *(const float4*)(phiB + (size_t)(kk + akr) * NPIX + p0 + acm);
        *(float4*)&Bs[bkr][bcn] =
            *(const float4*)(Cm + (size_t)(kk + bkr) * NCENT + q0 + bcn);
        *(float4*)&Bs[bkr + 4][bcn] =
            *(const float4*)(Cm + (size_t)(kk + bkr + 4) * NCENT + q0 + bcn);

        // prefetch next chunk's panels into cache (global_prefetch_b8)
        if (kk + KC < KDIM) {
            __builtin_prefetch(phiB + (size_t)(kk + KC + akr) * NPIX + p0 + acm, 0, 3);
            __builtin_prefetch(Cm + (size_t)(kk + KC + bkr) * NCENT + q0 + bcn, 0, 3);
            __builtin_prefetch(Cm + (size_t)(kk + KC + bkr + 4) * NCENT + q0 + bcn, 0, 3);
        }
        __syncthreads();

        // ---- two K=4 WMMA steps per chunk, fragments from LDS
        #pragma unroll
        for (int ks = 0; ks < KC; ks += 4) {
            int kb = ks + 2 * half;
            v2f a0 = { As[kb][am0],      As[kb + 1][am0]      };
            v2f a1 = { As[kb][am0 + 16], As[kb + 1][am0 + 16] };
            v2f b0 = { Bs[kb][bn0],      Bs[kb + 1][bn0]      };
            v2f b1 = { Bs[kb][bn0 + 16], Bs[kb + 1][bn0 + 16] };
            v2f b2 = { Bs[kb][bn0 + 32], Bs[kb + 1][bn0 + 32] };
            v2f b3 = { Bs[kb][bn0 + 48], Bs[kb + 1][bn0 + 48] };

            acc[0][0] = __builtin_amdgcn_wmma_f32_16x16x4_f32(false, a0, false, b0, (short)0, acc[0][0], false, false);
            acc[0][1] = __builtin_amdgcn_wmma_f32_16x16x4_f32(false, a0, false, b1, (short)0, acc[0][1], false, false);
            acc[0][2] = __builtin_amdgcn_wmma_f32_16x16x4_f32(false, a0, false, b2, (short)0, acc[0][2], false, false);
            acc[0][3] = __builtin_amdgcn_wmma_f32_16x16x4_f32(false, a0, false, b3, (short)0, acc[0][3], false, false);
            acc[1][0] = __builtin_amdgcn_wmma_f32_16x16x4_f32(false, a1, false, b0, (short)0, acc[1][0], false, false);
            acc[1][1] = __builtin_amdgcn_wmma_f32_16x16x4_f32(false, a1, false, b1, (short)0, acc[1][1], false, false);
            acc[1][2] = __builtin_amdgcn_wmma_f32_16x16x4_f32(false, a1, false, b2, (short)0, acc[1][2], false, false);
            acc[1][3] = __builtin_amdgcn_wmma_f32_16x16x4_f32(false, a1, false, b3, (short)0, acc[1][3], false, false);
        }
        __syncthreads();
    }

    // C/D layout: VGPR v holds M = v + 8*half, N = l16 (per 16x16 tile)
    #pragma unroll
    for (int t = 0; t < 2; ++t) {
        #pragma unroll
        for (int v = 0; v < 8; ++v) {
            int p = p0 + 32 * wm + 16 * t + 8 * half + v;
            float fp = featsB[p];
            #pragma unroll
            for (int u = 0; u < 4; ++u) {
                int q = q0 + 64 * wn + 16 * u + l16;
                float d2 = fp + cents[q] - 2.f * acc[t][u][v];
                dist[(size_t)(p - pc) * NCENT + q] = sqrtf(fmaxf(d2, 0.f));
            }
        }
    }
}

// ---------------- per-pixel top-200 smallest via in-LDS bitonic sort ---------
__global__ void topk_kernel(const float* __restrict__ dist, float* __restrict__ score,
                            int pc, int b) {
    __shared__ float sd[NCENT];
    int r = blockIdx.x;              // local row within chunk
    int tid = threadIdx.x;
    for (int i = tid; i < NCENT; i += blockDim.x)
        sd[i] = dist[(size_t)r * NCENT + i];
    __syncthreads();

    for (int k = 2; k <= NCENT; k <<= 1) {
        for (int j = k >> 1; j > 0; j >>= 1) {
            for (int i = tid; i < NCENT; i += blockDim.x) {
                int l = i ^ j;
                if (l > i) {
                    float a = sd[i], c = sd[l];
                    bool doswap = ((i & k) == 0) ? (a > c) : (a < c);
                    if (doswap) { sd[i] = c; sd[l] = a; }
                }
            }
            __syncthreads();
        }
    }

    int p = pc + r;
    for (int n = tid; n < NNEI; n += blockDim.x)
        score[((size_t)b * NNEI + n) * NPIX + p] = sd[n];   // ascending = top-k smallest
}

// ---------------------------------------------------------------------------
extern "C" void kernel_launch(void* const* d_in, const int* in_sizes, int n_in,
                              void* d_out, int out_size, void* d_ws, size_t ws_size,
                              hipStream_t stream) {
    (void)in_sizes; (void)n_in; (void)out_size; (void)ws_size;
    const float* p0in = (const float*)d_in[0];
    const float* p1in = (const float*)d_in[1];
    const float* p2in = (const float*)d_in[2];
    // d_in[3] = label (unused), d_in[4] = mask (unused)
    const float* w1 = (const float*)d_in[5];
    const float* b1 = (const float*)d_in[6];
    const float* w2 = (const float*)d_in[7];
    const float* b2 = (const float*)d_in[8];
    const float* w3 = (const float*)d_in[9];
    const float* b3 = (const float*)d_in[10];
    const float* Cm = (const float*)d_in[11];

    float* out   = (float*)d_out;
    float* out2  = out + (size_t)NB * NNEI * NPIX;       // PHI_P[:, :896]

    // workspace layout (floats)
    float* ws     = (float*)d_ws;
    float* pooled = ws;                                   //  4,194,304 (max level)
    float* convt  = ws + 4194304;                         //  4,194,304 (max level)
    float* phi    = ws + 8388608;                         // 29,360,128
    float* feats  = phi + (size_t)NB * KDIM * NPIX;       //     16,384
    float* cents  = feats + NB * NPIX;                    //      4,096
    float* dist   = cents + NCENT;                        //  2,097,152 (chunk)

    struct Lvl { const float* x; const float* w; const float* b; int C; int S; int off; };
    Lvl lv[3] = {
        { p0in, w1, b1,  256, 64,   0 },
        { p1in, w2, b2,  512, 32, 256 },
        { p2in, w3, b3, 1024, 16, 768 },
    };

    for (int L = 0; L < 3; ++L) {
        int C = lv[L].C, S = lv[L].S, SS = S * S;
        int tot = NB * C * SS;
        pool3_kernel<<<(tot + 255) / 256, 256, 0, stream>>>(lv[L].x, pooled, C, S);
        coordconv_kernel<<<dim3((SS + 255) / 256, C, NB), 256, 0, stream>>>(
            pooled, lv[L].w, lv[L].b, convt, C, S);
        resize_kernel<<<dim3(NPIX / 256, C, NB), 256, 0, stream>>>(
            convt, phi, out2, S, lv[L].off);
    }

    cents_kernel<<<NCENT / 256, 256, 0, stream>>>(Cm, cents);
    feats_kernel<<<dim3(NPIX / 256, NB), 256, 0, stream>>>(phi, feats);

    for (int b = 0; b < NB; ++b) {
        const float* phiB   = phi + (size_t)b * KDIM * NPIX;
        const float* featsB = feats + b * NPIX;
        for (int pc = 0; pc < NPIX; pc += PCHUNK) {
            gemm_dist_kernel<<<dim3(NCENT / 128, PCHUNK / 64), 128, 0, stream>>>(
                phiB, Cm, featsB, cents, dist, pc);
            topk_kernel<<<PCHUNK, 256, 0, stream>>>(dist, out, pc, b);
        }
    }
}